// GraphSAGE_69346541962022
// MI455X (gfx1250) — compile-verified
//
#include <hip/hip_runtime.h>
#include <hip/hip_bf16.h>

#define DIM 512  // feature dimension D (compile-time so /D, %D become shifts)

typedef float v2f __attribute__((ext_vector_type(2)));
typedef float v8f __attribute__((ext_vector_type(8)));

// ---------------------------------------------------------------------------
// zero a float buffer
__global__ void sage_zero_kernel(float* __restrict__ p, int n) {
    int i = blockIdx.x * blockDim.x + threadIdx.x;
    if (i < n) p[i] = 0.0f;
}

// ---------------------------------------------------------------------------
// degree: deg[dst[e]] += 1   (global_atomic_add_f32)
__global__ void sage_deg_kernel(const int* __restrict__ dst,
                                float* __restrict__ deg, int E) {
    int e = blockIdx.x * blockDim.x + threadIdx.x;
    if (e < E) atomicAdd(&deg[dst[e]], 1.0f);
}

// ---------------------------------------------------------------------------
// scatter-add: agg[dst[e], :] += h[src[e], :]
// one 128-thread block per edge; each thread moves 4 consecutive floats
// (float4 load from the gathered row, 4 f32 atomics into the segment sum)
__global__ void sage_scatter_kernel(const float* __restrict__ h,
                                    const int* __restrict__ src,
                                    const int* __restrict__ dst,
                                    float* __restrict__ agg) {
    int e = blockIdx.x;
    int s = src[e];
    int d = dst[e];
    const float4* row = (const float4*)(h + (size_t)s * DIM);
    float* outrow = agg + (size_t)d * DIM;
    int t = threadIdx.x;             // 0..127 -> 128*4 = 512 floats
    float4 v = row[t];
    atomicAdd(&outrow[t * 4 + 0], v.x);
    atomicAdd(&outrow[t * 4 + 1], v.y);
    atomicAdd(&outrow[t * 4 + 2], v.z);
    atomicAdd(&outrow[t * 4 + 3], v.w);
}

// ---------------------------------------------------------------------------
// h_neigh = agg / max(deg, 1)   (in place)
__global__ void sage_norm_kernel(float* __restrict__ agg,
                                 const float* __restrict__ deg, int n) {
    int i = blockIdx.x * blockDim.x + threadIdx.x;
    if (i < n) {
        int node = i / DIM;
        float dg = deg[node];
        agg[i] *= 1.0f / fmaxf(dg, 1.0f);
    }
}

// ---------------------------------------------------------------------------
// Fused dual GEMM + bias + ReLU, full fp32 via V_WMMA_F32_16X16X4_F32.
//   out = relu(A1 @ B1 + A2 @ B2 + bias)
// One wave32 per 16x16 output tile. N rows = ntm*16 (10000 = 625*16), D = 512.
//
// Per-lane operand layout (CDNA5 ISA 7.12.2, 32-bit A 16x4):
//   lanes 0-15 : rows M=lane,    VGPR0=K+0, VGPR1=K+1
//   lanes 16-31: rows M=lane-16, VGPR0=K+2, VGPR1=K+3
// B (4x16) mirrored: lanes 0-15 cols N=lane with rows K+0/K+1,
//   lanes 16-31 cols N=lane-16 with rows K+2/K+3.
// C/D f32 16x16: VGPR v -> M = v + (lane>=16 ? 8 : 0), N = lane&15.
__global__ void __launch_bounds__(128)
sage_gemm_kernel(const float* __restrict__ A1,   // self features  [N, D]
                 const float* __restrict__ A2,   // neigh features [N, D]
                 const float* __restrict__ B1,   // W_self  [D, D]
                 const float* __restrict__ B2,   // W_neigh [D, D]
                 const float* __restrict__ bias, // [D]
                 float* __restrict__ out) {      // [N, D]
    const int wave   = (blockIdx.x * blockDim.x + threadIdx.x) >> 5;
    const int lane   = threadIdx.x & 31;
    const int tile_m = wave >> 5;        // D/16 = 32 tiles along N-dim
    const int tile_n = wave & 31;

    const int l15  = lane & 15;
    const int hi   = lane >> 4;          // 0 for lanes 0-15, 1 for 16-31
    const int row  = tile_m * 16 + l15;  // A row this lane supplies
    const int col  = tile_n * 16 + l15;  // B/C column this lane supplies
    const int koff = hi * 2;             // K sub-offset per half-wave

    const float* a1p = A1 + (size_t)row * DIM + koff;
    const float* a2p = A2 + (size_t)row * DIM + koff;
    const float* b1p = B1 + (size_t)koff * DIM + col;
    const float* b2p = B2 + (size_t)koff * DIM + col;

    v8f c = {};
    #pragma unroll 4
    for (int k = 0; k < DIM; k += 4) {
        v2f a1 = *(const v2f*)(a1p + k);                 // 8B contiguous
        v2f b1; b1.x = b1p[(size_t)k * DIM];
                b1.y = b1p[(size_t)(k + 1) * DIM];
        c = __builtin_amdgcn_wmma_f32_16x16x4_f32(
                false, a1, false, b1, (short)0, c, false, false);

        v2f a2 = *(const v2f*)(a2p + k);
        v2f b2; b2.x = b2p[(size_t)k * DIM];
                b2.y = b2p[(size_t)(k + 1) * DIM];
        c = __builtin_amdgcn_wmma_f32_16x16x4_f32(
                false, a2, false, b2, (short)0, c, false, false);
    }

    const float bval = bias[col];
    const int   mrow = tile_m * 16 + hi * 8;
    #pragma unroll
    for (int v = 0; v < 8; v++) {
        float x = c[v] + bval;
        out[(size_t)(mrow + v) * DIM + col] = x > 0.0f ? x : 0.0f;
    }
}

// ---------------------------------------------------------------------------
extern "C" void kernel_launch(void* const* d_in, const int* in_sizes, int n_in,
                              void* d_out, int out_size, void* d_ws, size_t ws_size,
                              hipStream_t stream) {
    const float* feats = (const float*)d_in[0];
    const int*   src   = (const int*)  d_in[1];
    const int*   dst   = (const int*)  d_in[2];
    const float* Ws1   = (const float*)d_in[3];
    const float* Wn1   = (const float*)d_in[4];
    const float* b1    = (const float*)d_in[5];
    const float* Ws2   = (const float*)d_in[6];
    const float* Wn2   = (const float*)d_in[7];
    const float* b2    = (const float*)d_in[8];
    float*       out   = (float*)d_out;

    const int N  = in_sizes[0] / DIM;   // 10000
    const int E  = in_sizes[1];         // 160000
    const int ND = N * DIM;             // 5,120,000

    // workspace: [agg N*D][h1 N*D][deg N]
    float* agg = (float*)d_ws;
    float* h1  = agg + (size_t)ND;
    float* deg = h1 + (size_t)ND;

    const int ZB = 256;
    const int gemm_waves  = (N / 16) * (DIM / 16);   // 625 * 32 = 20000
    const int gemm_blocks = gemm_waves / 4;          // 4 waves / 128-thread block

    // ---- layer 1 aggregation ----
    sage_zero_kernel<<<(ND + ZB - 1) / ZB, ZB, 0, stream>>>(agg, ND);
    sage_zero_kernel<<<(N + ZB - 1) / ZB, ZB, 0, stream>>>(deg, N);
    sage_deg_kernel<<<(E + ZB - 1) / ZB, ZB, 0, stream>>>(dst, deg, E);
    sage_scatter_kernel<<<E, 128, 0, stream>>>(feats, src, dst, agg);
    sage_norm_kernel<<<(ND + ZB - 1) / ZB, ZB, 0, stream>>>(agg, deg, ND);

    // ---- layer 1 fused GEMM + bias + ReLU ----
    sage_gemm_kernel<<<gemm_blocks, 128, 0, stream>>>(feats, agg, Ws1, Wn1, b1, h1);

    // ---- layer 2 aggregation (deg unchanged) ----
    sage_zero_kernel<<<(ND + ZB - 1) / ZB, ZB, 0, stream>>>(agg, ND);
    sage_scatter_kernel<<<E, 128, 0, stream>>>(h1, src, dst, agg);
    sage_norm_kernel<<<(ND + ZB - 1) / ZB, ZB, 0, stream>>>(agg, deg, ND);

    // ---- layer 2 fused GEMM + bias + ReLU -> d_out ----
    sage_gemm_kernel<<<gemm_blocks, 128, 0, stream>>>(h1, agg, Ws2, Wn2, b2, out);
}